// PowerBasket_82832739271248
// MI455X (gfx1250) — compile-verified
//
#include <hip/hip_runtime.h>
#include <hip/hip_bf16.h>

namespace {
constexpr int kNItems   = 100000;
constexpr int kNPrices  = 10000;
constexpr int kNBaskets = 100000;
constexpr int kL        = 20;   // items per basket
constexpr int kK        = 16;   // baskets per item/price
constexpr int kD        = 64;   // embedding dim

constexpr int kWavesPerBlock = 8;
constexpr int kBlock         = kWavesPerBlock * 32;
constexpr int kBPad          = 68; // padded row stride for B tile in LDS (conflict-free b64 A-frag reads)
// Paired W1 layout: pair index (kc*2+half)*80 + col, in float2 units.
// Half-wave stride 80 float2 = 160 DWORDs == 32 (mod 64) banks -> lanes 16-31
// hit the complementary 32 banks of lanes 0-15: conflict-free ds_load_b64.
constexpr int kWHalfStride = 80;
}

typedef float v2f __attribute__((ext_vector_type(2)));
typedef float v8f __attribute__((ext_vector_type(8)));

__device__ __forceinline__ float fast_tanh(float x) {
#if __has_builtin(__builtin_amdgcn_tanhf)
  return __builtin_amdgcn_tanhf(x);   // gfx1250 v_tanh_f32 (TRANS32)
#else
  return tanhf(x);
#endif
}

// ---------------------------------------------------------------------------
// Kernel 1: basket_emb[b][:] = tanh(mean_l table[ids[b][l]][:])
// One wave (32 lanes) per basket; lane owns a float2 (2 of 64 columns).
// ---------------------------------------------------------------------------
__global__ __launch_bounds__(kBlock) void pb_basket_emb(
    const float* __restrict__ table,
    const int*   __restrict__ ids,
    float*       __restrict__ out,
    int nBaskets)
{
  const int wave = (int)(blockIdx.x * (unsigned)kWavesPerBlock + (threadIdx.x >> 5));
  const int lane = threadIdx.x & 31;
  const int b    = (wave < nBaskets) ? wave : 0;

  int idxReg = 0;
  if (lane < kL) idxReg = ids[(size_t)b * kL + lane];

  float ax = 0.f, ay = 0.f;
#pragma unroll
  for (int l = 0; l < kL; ++l) {
    const int idx  = __shfl(idxReg, l, 32);
    const float2 v = *(const float2*)(table + (size_t)idx * kD + 2 * lane);
    ax += v.x; ay += v.y;
  }
  const float inv = 1.0f / (float)kL;
  float2 r;
  r.x = fast_tanh(ax * inv);
  r.y = fast_tanh(ay * inv);
  if (wave < nBaskets)
    *(float2*)(out + (size_t)wave * kD + 2 * lane) = r;
}

// ---------------------------------------------------------------------------
// Kernel 2: attention pooling update. One wave per item.
//   B (16x64) gathered to LDS; H = tanh(B*W1 + b1) via V_WMMA_F32_16X16X4_F32
//   (M=16, N=64 in 4 tiles, K=64 in 16 steps of 4); s = H*w2; masked softmax;
//   out = (c<2) ? B[0] : emb + a^T B.
// ---------------------------------------------------------------------------
__global__ __launch_bounds__(kBlock) void pb_attn_update(
    const float* __restrict__ table,       // (n, 64) original embeddings
    const int*   __restrict__ id2basket,   // (n, 16)
    const int*   __restrict__ counts,      // (n,)
    const float* __restrict__ basket_emb,  // (N_BASKETS, 64)
    const float* __restrict__ W1,          // (64, 64) row-major [d][e]
    const float* __restrict__ b1,          // (64,)
    const float* __restrict__ w2,          // (64,)
    float*       __restrict__ out,         // (n, 64)
    int n)
{
  __shared__ v2f   sW1p[kK * 2 * kWHalfStride];  // paired B-fragment layout, 20.5 KB
  __shared__ float sb1[kD];
  __shared__ float sw2[kD];
  __shared__ float sB[kWavesPerBlock][kK][kBPad];

  // Stage W1 pre-paired in WMMA B-operand layout:
  //   sW1p[(kc*2+half)*80 + col] = { W1[4kc+2half][col], W1[4kc+2half+1][col] }
  for (int p = threadIdx.x; p < kD * kK * 2 / 2 * 2; p += kBlock) { // 2048 pairs
    const int col = p & 63;
    const int kh  = p >> 6;        // kc*2 + half
    const int row = kh * 2;        // 4kc + 2half
    v2f v;
    v.x = W1[row * kD + col];
    v.y = W1[(row + 1) * kD + col];
    sW1p[kh * kWHalfStride + col] = v;
  }
  if (threadIdx.x < kD) {
    sb1[threadIdx.x] = b1[threadIdx.x];
    sw2[threadIdx.x] = w2[threadIdx.x];
  }
  __syncthreads();

  const int w    = threadIdx.x >> 5;
  const int lane = threadIdx.x & 31;
  const int item = (int)(blockIdx.x * (unsigned)kWavesPerBlock + w);
  const int it   = (item < n) ? item : 0;   // clamp; no early return (EXEC all-1 for WMMA)

  const int c = counts[it] + 1;             // reference: c = count + 1, c in [1,16]

  int bidReg = 0;
  if (lane < kK) bidReg = id2basket[(size_t)it * kK + lane];

  // Gather B tile into per-wave LDS region (coalesced 256B rows).
  const int d0 = 2 * lane;
#pragma unroll
  for (int k = 0; k < kK; ++k) {
    const int bid  = __shfl(bidReg, k, 32);
    const float2 v = *(const float2*)(basket_emb + (size_t)bid * kD + d0);
    sB[w][k][d0]     = v.x;
    sB[w][k][d0 + 1] = v.y;
  }
  // LDS pipeline keeps same-wave DS ops in order; just stop compiler reordering.
  asm volatile("" ::: "memory");

  const int r    = lane & 15;        // A row / C column-within-tile
  const int half = lane >> 4;        // K split: 0 -> K+0/K+1, 1 -> K+2/K+3
  const int cb   = half << 1;

  // C accumulators initialized with broadcast bias (C[j]: N = nt*16 + r for all j).
  v8f cfrag[4];
#pragma unroll
  for (int nt = 0; nt < 4; ++nt) {
    const float bv = sb1[nt * 16 + r];
    cfrag[nt] = (v8f){bv, bv, bv, bv, bv, bv, bv, bv};
  }

  // 64 x V_WMMA_F32_16X16X4_F32: accumulate over K=64 in 16 chunks of 4.
#pragma unroll
  for (int kc = 0; kc < kK; ++kc) {
    // A fragment: one conflict-free ds_load_b64 per lane.
    const v2f afrag = *(const v2f*)&sB[w][r][4 * kc + cb];
    const int wbase = (kc * 2 + half) * kWHalfStride + r;
#pragma unroll
    for (int nt = 0; nt < 4; ++nt) {
      // B fragment: pre-paired, one conflict-free ds_load_b64, no repack movs.
      const v2f bfrag = sW1p[wbase + nt * 16];
      cfrag[nt] = __builtin_amdgcn_wmma_f32_16x16x4_f32(
          false, afrag, false, bfrag, (short)0, cfrag[nt], false, false);
    }
  }

  // H = tanh(pre-activation); lane<16 holds rows 0..7 (VGPR j -> M=j),
  // lane>=16 holds rows 8..15 (M=j+8), column e = nt*16 + r.
  float H[4][8];
#pragma unroll
  for (int nt = 0; nt < 4; ++nt)
#pragma unroll
    for (int j = 0; j < 8; ++j)
      H[nt][j] = fast_tanh(cfrag[nt][j]);

  // Scores s[m] = sum_e H[m][e]*w2[e]: per-lane partial then 16-lane reduction.
  float p[8];
#pragma unroll
  for (int j = 0; j < 8; ++j) {
    float acc = 0.f;
#pragma unroll
    for (int nt = 0; nt < 4; ++nt)
      acc += H[nt][j] * sw2[nt * 16 + r];
    p[j] = acc;
  }
#pragma unroll
  for (int m = 1; m <= 8; m <<= 1)
#pragma unroll
    for (int j = 0; j < 8; ++j)
      p[j] += __shfl_xor(p[j], m, 32);

  // Masked softmax over 16 rows (rows base..base+7 live in this half-wave).
  const int base = half << 3;
  float sc[8];
#pragma unroll
  for (int j = 0; j < 8; ++j)
    sc[j] = (base + j < c) ? p[j] : -1.0e9f;

  float mx = sc[0];
#pragma unroll
  for (int j = 1; j < 8; ++j) mx = fmaxf(mx, sc[j]);
  mx = fmaxf(mx, __shfl_xor(mx, 16, 32));

  float e8[8];
  float sum = 0.f;
#pragma unroll
  for (int j = 0; j < 8; ++j) { e8[j] = __expf(sc[j] - mx); sum += e8[j]; }
  sum += __shfl_xor(sum, 16, 32);
  const float invs = 1.0f / sum;

  // Full a[0..15] per lane via cross-half exchange.
  float a[16];
#pragma unroll
  for (int j = 0; j < 8; ++j) {
    const float own = e8[j] * invs;
    const float oth = __shfl_xor(own, 16, 32);
    if (lane < 16) { a[j] = own;  a[8 + j] = oth; }
    else           { a[j] = oth;  a[8 + j] = own; }
  }

  // pooled[d] = sum_k a[k]*B[k][d]; lane owns columns d0, d0+1.
  float px = 0.f, py = 0.f;
#pragma unroll
  for (int k = 0; k < kK; ++k) {
    const float ak = a[k];
    px += ak * sB[w][k][d0];
    py += ak * sB[w][k][d0 + 1];
  }

  float2 res;
  if (c < 2) {
    res.x = sB[w][0][d0];
    res.y = sB[w][0][d0 + 1];
  } else {
    const float2 t = *(const float2*)(table + (size_t)it * kD + d0);
    res.x = t.x + px;
    res.y = t.y + py;
  }
  if (item < n)
    *(float2*)(out + (size_t)item * kD + d0) = res;
}

// ---------------------------------------------------------------------------
// Launch. d_ws needs 2 * N_BASKETS * 64 * 4 B = 51.2 MB for the two basket-
// embedding tables (fully L2-resident on MI455X's 192 MB L2).
// ---------------------------------------------------------------------------
extern "C" void kernel_launch(void* const* d_in, const int* in_sizes, int n_in,
                              void* d_out, int out_size, void* d_ws, size_t ws_size,
                              hipStream_t stream) {
  const float* item_emb      = (const float*)d_in[0];
  const float* price_emb     = (const float*)d_in[1];
  const int*   basket_items  = (const int*)d_in[2];
  const int*   basket_prices = (const int*)d_in[3];
  const int*   item2basket   = (const int*)d_in[4];
  const int*   item_count    = (const int*)d_in[5];
  const int*   price2basket  = (const int*)d_in[6];
  const int*   price_count   = (const int*)d_in[7];
  const float* Wi1 = (const float*)d_in[8];
  const float* bi1 = (const float*)d_in[9];
  const float* wi2 = (const float*)d_in[10];
  const float* Wp1 = (const float*)d_in[11];
  const float* bp1 = (const float*)d_in[12];
  const float* wp2 = (const float*)d_in[13];

  float* out = (float*)d_out;
  float* bei = (float*)d_ws;                       // (N_BASKETS, 64)
  float* bep = bei + (size_t)kNBaskets * kD;       // (N_BASKETS, 64)

  const int gB = (kNBaskets + kWavesPerBlock - 1) / kWavesPerBlock;   // 12500
  pb_basket_emb<<<gB, kBlock, 0, stream>>>(item_emb,  basket_items,  bei, kNBaskets);
  pb_basket_emb<<<gB, kBlock, 0, stream>>>(price_emb, basket_prices, bep, kNBaskets);

  const int gI = (kNItems  + kWavesPerBlock - 1) / kWavesPerBlock;    // 12500
  const int gP = (kNPrices + kWavesPerBlock - 1) / kWavesPerBlock;    // 1250
  pb_attn_update<<<gI, kBlock, 0, stream>>>(item_emb, item2basket, item_count,
                                            bei, Wi1, bi1, wi2, out, kNItems);
  pb_attn_update<<<gP, kBlock, 0, stream>>>(price_emb, price2basket, price_count,
                                            bep, Wp1, bp1, wp2,
                                            out + (size_t)kNItems * kD, kNPrices);
}